// Pool4MLP_attention_71227737637096
// MI455X (gfx1250) — compile-verified
//
#include <hip/hip_runtime.h>
#include <hip/hip_bf16.h>

// Problem constants (match the reference)
#define BB 2
#define CC 64
#define NN 64
#define HH 128
#define WW 128
#define HW (HH * WW)          // 16384
#define PP 32
#define KK 8
#define NTILES (HW / 16)      // 1024 column tiles of 16 in pass 3

typedef float v2f __attribute__((ext_vector_type(2)));
typedef float v8f __attribute__((ext_vector_type(8)));

// ---------------------------------------------------------------------------
// Pass 1: per-(b,c,n) sum over H*W.  Reads 512 MB — the bandwidth-dominant
// pass (~22 us roofline at 23.3 TB/s).  One block per slab, float4 streaming.
// ---------------------------------------------------------------------------
__global__ void slab_sum_kernel(const float* __restrict__ x,
                                float* __restrict__ sums) {
    __shared__ float red[256];
    const int slab = blockIdx.x;                       // flat (b*C + c)*N + n
    const float4* p = (const float4*)(x + (size_t)slab * HW);
    float s = 0.f;
#pragma unroll
    for (int i = 0; i < 16; ++i) {                     // 256 thr * 16 * 4 = 16384
        float4 v = p[threadIdx.x + 256 * i];
        s += (v.x + v.y) + (v.z + v.w);
    }
    red[threadIdx.x] = s;
    __syncthreads();
    for (int off = 128; off > 0; off >>= 1) {
        if (threadIdx.x < off) red[threadIdx.x] += red[threadIdx.x + off];
        __syncthreads();
    }
    if (threadIdx.x == 0) sums[slab] = red[0];
}

// ---------------------------------------------------------------------------
// Pass 2: pooled features -> linear score -> softmax over N -> stable top-8.
// Tiny: 2 blocks x 64 threads.  Rank-count top-k reproduces jax.lax.top_k
// ordering (descending values, ties broken by lower index).
// ---------------------------------------------------------------------------
__global__ void score_topk_kernel(const float* __restrict__ sums,
                                  const float* __restrict__ w_score,
                                  const float* __restrict__ b_score,
                                  int* __restrict__ topk_idx,
                                  float* __restrict__ topk_val) {
    __shared__ float sc[NN];
    __shared__ float ex[NN];
    __shared__ float wv[NN];
    const int b = blockIdx.x;
    const int n = threadIdx.x;                         // 64 threads = N slices

    float acc = 0.f;
#pragma unroll
    for (int pch = 0; pch < PP; ++pch) {
        float s0 = sums[((size_t)b * CC + 2 * pch)     * NN + n];
        float s1 = sums[((size_t)b * CC + 2 * pch + 1) * NN + n];
        // adaptive pool (avg of channel pair) then mean over H*W
        acc += (0.5f * (s0 + s1) * (1.0f / (float)HW)) * w_score[pch];
    }
    acc += b_score[0];
    sc[n] = acc;
    __syncthreads();

    float mx = -INFINITY;
    for (int m = 0; m < NN; ++m) mx = fmaxf(mx, sc[m]);
    float e = expf(acc - mx);
    ex[n] = e;
    __syncthreads();

    float se = 0.f;
    for (int m = 0; m < NN; ++m) se += ex[m];
    float wgt = e / se;
    wv[n] = wgt;
    __syncthreads();

    int rank = 0;
    for (int m = 0; m < NN; ++m) {
        float wm = wv[m];
        rank += (wm > wgt) || (wm == wgt && m < n);    // stable descending rank
    }
    if (rank < KK) {
        topk_idx[b * KK + rank] = n;
        topk_val[b * KK + rank] = wgt;
    }
}

// ---------------------------------------------------------------------------
// Pass 3: fused gather + channel-pool + scale via V_WMMA_F32_16X16X4_F32.
//   OUT(32 x 16384) = pool_mat(32x64) @ X(64x16384) * topk_val, per (b,k).
// One wave owns one 16(p) x 16(hw) tile; K=64 in 16 accumulating WMMA steps.
// A (pool tile) uses the ISA-documented f32 16x4 layout:
//   lanes 0-15: V0=K0,V1=K1 ; lanes 16-31: V0=K2,V1=K3 ; row M = lane%16.
// B (data tile, 4x16) assumed mirror layout of A ("B rows striped across
// lanes within one VGPR"): lanes 0-15 hold rows K0(V0)/K1(V1) at col N=lane,
// lanes 16-31 hold rows K2/K3.
// D uses the documented C/D layout: col N = lane%16, row M = vgpr + 8*(lane/16).
// Grid divides exactly -> EXEC is all-1s around every WMMA (ISA requirement).
// ---------------------------------------------------------------------------
__global__ void gather_pool_wmma_kernel(const float* __restrict__ x,
                                        const int* __restrict__ topk_idx,
                                        const float* __restrict__ topk_val,
                                        float* __restrict__ out) {
    const int wave = blockIdx.x * (blockDim.x >> 5) + (threadIdx.x >> 5);
    const int lane = threadIdx.x & 31;
    const int lan16 = lane & 15;
    const int khalf = lane >> 4;

    // decode: wave = (((b*K + k)*2 + mtile)*NTILES + ntile)
    const int ntile = wave & (NTILES - 1);
    int rest = wave >> 10;
    const int mtile = rest & 1;  rest >>= 1;
    const int k     = rest & (KK - 1);  rest >>= 3;
    const int b     = rest;

    const int   nslice = topk_idx[b * KK + k];     // uniform per wave
    const float val    = topk_val[b * KK + k];

    const int hw0 = ntile * 16;
    const int p0  = mtile * 16;
    const size_t xbase = ((size_t)b * CC * NN + nslice) * HW;  // c = 0 plane

    v8f acc = {};
#pragma unroll
    for (int kk = 0; kk < 16; ++kk) {
        const int c0  = kk * 4;
        const int ck  = c0 + 2 * khalf;            // this lane-half's K rows
        // A: pool matrix tile (exact 0.5 weights, generated inline)
        const int pa = p0 + lan16;
        v2f a;
        a.x = (((ck    ) >> 1) == pa) ? 0.5f : 0.0f;
        a.y = (((ck + 1) >> 1) == pa) ? 0.5f : 0.0f;
        // B: X tile — column hw0+lan16 of channel rows ck, ck+1
        const size_t col = xbase + (size_t)(hw0 + lan16);
        v2f bm;
        bm.x = x[col + (size_t)(ck    ) * ((size_t)NN * HW)];
        bm.y = x[col + (size_t)(ck + 1) * ((size_t)NN * HW)];
        acc = __builtin_amdgcn_wmma_f32_16x16x4_f32(
            /*neg_a=*/false, a, /*neg_b=*/false, bm,
            /*c_mod=*/(short)0, acc, /*reuse_a=*/false, /*reuse_b=*/false);
    }

    // scatter D (scaled by softmax weight) to out[b, p, k, hw]
#pragma unroll
    for (int r = 0; r < 8; ++r) {
        const int p = p0 + r + 8 * khalf;
        const size_t oidx =
            (((size_t)b * PP + p) * KK + k) * HW + (size_t)(hw0 + lan16);
        out[oidx] = acc[r] * val;
    }
}

// ---------------------------------------------------------------------------
extern "C" void kernel_launch(void* const* d_in, const int* in_sizes, int n_in,
                              void* d_out, int out_size, void* d_ws, size_t ws_size,
                              hipStream_t stream) {
    const float* x       = (const float*)d_in[0];   // [B,C,N,H,W] f32
    const float* w_score = (const float*)d_in[1];   // [1,P] f32 (flat 32)
    const float* b_score = (const float*)d_in[2];   // [1]   f32
    float* out = (float*)d_out;                     // [B,P,K,H,W] f32

    // workspace layout: sums [B*C*N] f32 | topk_idx [B*K] i32 | topk_val [B*K] f32
    float* sums = (float*)d_ws;
    int*   tidx = (int*)((char*)d_ws + (size_t)BB * CC * NN * sizeof(float));
    float* tval = (float*)((char*)d_ws + (size_t)BB * CC * NN * sizeof(float)
                                       + (size_t)BB * KK * sizeof(int));

    // Pass 1: 8192 slabs, each 64 KB of floats reduced by one 256-thread block
    slab_sum_kernel<<<BB * CC * NN, 256, 0, stream>>>(x, sums);
    // Pass 2: one block per batch
    score_topk_kernel<<<BB, NN, 0, stream>>>(sums, w_score, b_score, tidx, tval);
    // Pass 3: B*K*2*NTILES = 32768 waves, 8 waves (256 threads) per block
    gather_pool_wmma_kernel<<<(BB * KK * 2 * NTILES) / 8, 256, 0, stream>>>(
        x, tidx, tval, out);
}